// ParamVectorPool_63556926046520
// MI455X (gfx1250) — compile-verified
//
#include <hip/hip_runtime.h>
#include <math.h>

// Dims from the reference
#define LDIM 7
#define SDIM 200
#define SEDIM 240
#define DDIM 32
#define KSEL 3
#define BDIM 512

#define NTILE ((SDIM + 15) / 16)   // 13 column tiles of the router GEMM
#define SPAD  (NTILE * 16)         // 208 (padded S for LDS)

typedef __attribute__((ext_vector_type(2))) float v2f;
typedef __attribute__((ext_vector_type(8))) float v8f;

// ---------------------------------------------------------------------------
// Kernel 1: router. One wave (32 lanes) computes logits for a 16-token tile
// against all S clusters with V_WMMA_F32_16X16X4_F32, then top-3 + softmax.
// grid = (B/16, L), block = 32.
// ---------------------------------------------------------------------------
__global__ __launch_bounds__(32) void router_kernel(
    const float* __restrict__ x,        // [L,B,D]
    const float* __restrict__ vc,       // [L,S,D]
    int* __restrict__ top_idx,          // [L,B,K]
    float* __restrict__ probs)          // [L,B,K]
{
    __shared__ float lds[16][SPAD];

    const int lane  = threadIdx.x;          // 0..31
    const int btile = blockIdx.x;           // 0..B/16-1
    const int l     = blockIdx.y;           // 0..L-1
    const int b0    = btile * 16;

    const int mrow  = lane & 15;            // M index within tile
    const int khalf = (lane >> 4) * 2;      // lanes 0-15 -> K={0,1}, 16-31 -> K={2,3}

    // A fragments: x tile [16 tokens x 32 dims], 8 k-steps of K=4.
    const float* xrow = x + ((size_t)l * BDIM + b0 + mrow) * DDIM;
    v2f afrag[8];
#pragma unroll
    for (int kk = 0; kk < 8; ++kk) {
        const int k0 = kk * 4 + khalf;
        afrag[kk].x = xrow[k0 + 0];
        afrag[kk].y = xrow[k0 + 1];
    }

    // Sweep cluster tiles (N dimension), accumulate 16x16 f32 per tile.
    for (int nt = 0; nt < NTILE; ++nt) {
        const int   n     = nt * 16 + mrow;
        const float scale = (n < SDIM) ? 1.0f : 0.0f;   // zero-pad tail tile
        const int   nc    = (n < SDIM) ? n : 0;
        const float* crow = vc + ((size_t)l * SDIM + nc) * DDIM;

        v8f c = {};
#pragma unroll
        for (int kk = 0; kk < 8; ++kk) {
            const int k0 = kk * 4 + khalf;
            v2f bfrag;
            bfrag.x = crow[k0 + 0] * scale;   // B[k][n] = centroid[n][k]
            bfrag.y = crow[k0 + 1] * scale;
            c = __builtin_amdgcn_wmma_f32_16x16x4_f32(
                    false, afrag[kk], false, bfrag, (short)0, c, false, false);
        }

        // C layout: VGPR j -> (M=j, N=lane) for lanes 0-15, (M=j+8) for 16-31.
        const int row = (lane < 16) ? 0 : 8;
        const int col = nt * 16 + mrow;
#pragma unroll
        for (int j = 0; j < 8; ++j)
            lds[row + j][col] = c[j];
    }
    __syncthreads();

    // Lanes 0..15: serial top-3 over S=200 (strict '>' == top_k tie-breaking),
    // then softmax over the 3 kept logits.
    if (lane < 16) {
        float v0 = -__builtin_inff(), v1 = v0, v2 = v0;
        int   i0 = 0, i1 = 0, i2 = 0;
        for (int s = 0; s < SDIM; ++s) {
            const float v = lds[lane][s];
            if (v > v0)      { v2 = v1; i2 = i1; v1 = v0; i1 = i0; v0 = v; i0 = s; }
            else if (v > v1) { v2 = v1; i2 = i1; v1 = v;  i1 = s; }
            else if (v > v2) { v2 = v;  i2 = s; }
        }
        const float e0 = __expf(v0 - v0);
        const float e1 = __expf(v1 - v0);
        const float e2 = __expf(v2 - v0);
        const float inv = 1.0f / (e0 + e1 + e2);

        const size_t base = ((size_t)l * BDIM + b0 + lane) * KSEL;
        top_idx[base + 0] = i0;  probs[base + 0] = e0 * inv;
        top_idx[base + 1] = i1;  probs[base + 1] = e1 * inv;
        top_idx[base + 2] = i2;  probs[base + 2] = e2 * inv;
    }
}

// ---------------------------------------------------------------------------
// Kernel 2: gather + weighted matvec + top-k mix. One wave per (l,b) token;
// lane = output dim d. vec reads are lane-contiguous (128B/wave per e).
// block = 8 waves.
// ---------------------------------------------------------------------------
#define WPB 8

__global__ __launch_bounds__(WPB * 32) void compose_kernel(
    const float* __restrict__ vec,      // [L,S,SE,D]
    const float* __restrict__ gain,     // [L,S,SE,2]
    const int*   __restrict__ blk,      // [L,S,SE,2]
    const int*   __restrict__ top_idx,  // [L,B,K]
    const float* __restrict__ probs,    // [L,B,K]
    float* __restrict__ out)            // [L,B,D]
{
    __shared__ float wls[WPB][SEDIM];

    const int lane = threadIdx.x & 31;
    const int wid  = threadIdx.x >> 5;
    const int t    = blockIdx.x * WPB + wid;       // flat (l*B + b)
    if (t >= LDIM * BDIM) return;

    const int l = t / BDIM;
    float acc = 0.0f;

#pragma unroll
    for (int k = 0; k < KSEL; ++k) {
        const int   s = top_idx[(size_t)t * KSEL + k];
        const float p = probs[(size_t)t * KSEL + k];

        const size_t slab = ((size_t)l * SDIM + s) * SEDIM;

        // Capacity-slot weights into LDS: sigmoid(gain) masked by blk != 0.
        for (int e = lane; e < SEDIM; e += 32) {
            const size_t gi = (slab + e) * 2;       // [...,0] component
            const float  g  = gain[gi];
            const int    bi = blk[gi];
            wls[wid][e] = (bi != 0) ? (1.0f / (1.0f + __expf(-g))) : 0.0f;
        }
        // Same-wave LDS ops are in-order; no barrier needed.

        // Matvec: comp[d] = sum_e w[e] * vec[l,s,e,d], lane = d.
        float a = 0.0f;
        const float* vp = vec + slab * DDIM + lane;
#pragma unroll 8
        for (int e = 0; e < SEDIM; ++e)
            a = fmaf(wls[wid][e], vp[(size_t)e * DDIM], a);

        acc = fmaf(p, a, acc);
    }

    out[(size_t)t * DDIM + lane] = acc;
}

// ---------------------------------------------------------------------------
extern "C" void kernel_launch(void* const* d_in, const int* in_sizes, int n_in,
                              void* d_out, int out_size, void* d_ws, size_t ws_size,
                              hipStream_t stream)
{
    const float* x    = (const float*)d_in[0];   // [L,B,D]
    const float* vc   = (const float*)d_in[1];   // [L,S,D]
    const float* vec  = (const float*)d_in[2];   // [L,S,SE,D]
    const float* gain = (const float*)d_in[3];   // [L,S,SE,2]
    const int*   blk  = (const int*)d_in[4];     // [L,S,SE,2]
    float*       out  = (float*)d_out;           // [L,B,D]

    // Workspace: top-k indices then probs.
    int*   top_idx = (int*)d_ws;
    float* probs   = (float*)((char*)d_ws + sizeof(int) * (size_t)LDIM * BDIM * KSEL);

    dim3 g1(BDIM / 16, LDIM);
    router_kernel<<<g1, 32, 0, stream>>>(x, vc, top_idx, probs);

    const int nwaves = LDIM * BDIM;
    compose_kernel<<<(nwaves + WPB - 1) / WPB, WPB * 32, 0, stream>>>(
        vec, gain, blk, top_idx, probs, out);
}